// CapsuleLayer_64570538328445
// MI455X (gfx1250) — compile-verified
//
#include <hip/hip_runtime.h>
#include <math.h>

// CapsuleLayer dynamic routing, fused for MI455X (gfx1250).
// u_hat (188.7 MB) is never materialized; all contractions are refactored into
// fp32 WMMA GEMMs over the L2-resident inputs (9.4 MB) + W (5.9 MB).
// This revision amortizes the A-fragment across all J=10 output capsules:
// 10 accumulators live in registers, so each A load feeds 10 wmma ops.

typedef __attribute__((ext_vector_type(2))) float v2f;
typedef __attribute__((ext_vector_type(8))) float v8f;

#define CAP_B   256
#define CAP_I   1152
#define CAP_DIN 8
#define CAP_J   10
#define CAP_D   16
#define CAP_IK  (CAP_I * CAP_DIN)   // 9216 = GEMM1 K dim
#define CAP_N   (CAP_J * CAP_D)     // 160  = N dim (16-wide tile n == capsule j)
#define CAP_EPS 1e-7f

#define ICHUNK   16
#define NCHUNKS  (CAP_I / ICHUNK)   // 72-way K-split for GEMM1 occupancy
#define BCHUNK   64
#define NBCHUNKS (CAP_B / BCHUNK)   // 4-way batch split for GEMM2

// ---------------------------------------------------------------------------
// softmax over J per input capsule: c[i,:] = softmax(blog[i,:])
// ---------------------------------------------------------------------------
__global__ void softmax_k(const float* __restrict__ blog, float* __restrict__ c) {
    int i = blockIdx.x * blockDim.x + threadIdx.x;
    if (i >= CAP_I) return;
    float row[CAP_J];
    float mx = -__builtin_inff();
    #pragma unroll
    for (int j = 0; j < CAP_J; ++j) { row[j] = blog[i * CAP_J + j]; mx = fmaxf(mx, row[j]); }
    float sum = 0.f;
    #pragma unroll
    for (int j = 0; j < CAP_J; ++j) { row[j] = __expf(row[j] - mx); sum += row[j]; }
    float inv = 1.0f / sum;
    #pragma unroll
    for (int j = 0; j < CAP_J; ++j) c[i * CAP_J + j] = row[j] * inv;
}

// ---------------------------------------------------------------------------
// GEMM1: s[b, j*16+d] += sum_{(i,k) in chunk} X'[b,(i,k)] * c[i,j]*W[i,j,k,d]
// One wave per block; M-tile = 16 batches, N = all 10 capsules (10 f32 accs),
// K split into 72 chunks of 16 input capsules. A-fragment loaded once per
// K-step and reused by 10 consecutive wmma ops.
// Fragment layouts per CDNA5 ISA 7.12.2:
//   A(16x4): lane L holds m=L%16, reg r holds k = r + 2*(L/16)
//   B(4x16): lane L holds n=L%16, reg r holds k = r + 2*(L/16)
//   C/D:     lane L holds n=L%16, reg r holds m = r + 8*(L/16)
// ---------------------------------------------------------------------------
__global__ __launch_bounds__(32) void s_gemm_k(const float* __restrict__ X,
                                               const float* __restrict__ W,
                                               const float* __restrict__ c,
                                               float* __restrict__ s) {
    int bid   = blockIdx.x;
    int chunk = bid % NCHUNKS;
    int mt    = bid / NCHUNKS;              // 0..15 batch tile
    int lane  = threadIdx.x;
    int half  = lane >> 4;
    int ln    = lane & 15;
    int b0    = mt * 16;

    v8f acc[CAP_J] = {};
    int i0 = chunk * ICHUNK;
    for (int ii = 0; ii < ICHUNK; ++ii) {
        int i = i0 + ii;
        const float* cp = c + i * CAP_J;                              // uniform -> s_load
        const float* wb = W + (size_t)i * CAP_J * (CAP_DIN * CAP_D);  // 10 x (8x16) blocks
        const float* xp = X + (size_t)(b0 + ln) * CAP_IK + i * CAP_DIN; // A row m=ln
        #pragma unroll
        for (int ks = 0; ks < 2; ++ks) {      // K=8 per capsule -> two K=4 steps
            int kb = 4 * ks + 2 * half;
            v2f a;
            a.x = xp[kb + 0];
            a.y = xp[kb + 1];
            #pragma unroll
            for (int j = 0; j < CAP_J; ++j) { // reuse A across all output capsules
                float cij = cp[j];
                const float* wp = wb + j * (CAP_DIN * CAP_D);
                v2f bb;
                bb.x = cij * wp[(kb + 0) * CAP_D + ln];
                bb.y = cij * wp[(kb + 1) * CAP_D + ln];
                acc[j] = __builtin_amdgcn_wmma_f32_16x16x4_f32(
                             false, a, false, bb, (short)0, acc[j], false, false);
            }
        }
    }
    // accumulate partial K-chunks into s with fp32 global atomics
    #pragma unroll
    for (int j = 0; j < CAP_J; ++j) {
        #pragma unroll
        for (int r = 0; r < 8; ++r) {
            int m = r + 8 * half;
            atomicAdd(&s[(size_t)(b0 + m) * CAP_N + j * CAP_D + ln], acc[j][r]);
        }
    }
}

// ---------------------------------------------------------------------------
// squash: v = (|s|^2 / (1+|s|^2)) * s / sqrt(|s|^2 + eps), per (b,j) vector
// ---------------------------------------------------------------------------
__global__ void squash_k(const float* __restrict__ s, float* __restrict__ v) {
    int idx = blockIdx.x * blockDim.x + threadIdx.x;   // over B*J
    if (idx >= CAP_B * CAP_J) return;
    const float* sp = s + (size_t)idx * CAP_D;
    float s2 = 0.f;
    #pragma unroll
    for (int d = 0; d < CAP_D; ++d) { float t = sp[d]; s2 += t * t; }
    float scale = (s2 / (1.0f + s2)) * __frsqrt_rn(s2 + CAP_EPS);
    float* vp = v + (size_t)idx * CAP_D;
    #pragma unroll
    for (int d = 0; d < CAP_D; ++d) vp[d] = sp[d] * scale;
}

// ---------------------------------------------------------------------------
// GEMM2 + fused epilogue: T = X'^T(9216x256) @ V(256x160), tile (16 ik x 16 d),
// then a[i,j] = sum_{k,d} W[i,j,k,d] * T[(i,k),(j,d)], atomically added into
// the routing logits. One wave per (ik-tile, batch-chunk); all 10 j in regs so
// the strided A-column fragment is loaded once per K-step.
// ---------------------------------------------------------------------------
__global__ __launch_bounds__(32) void agree_gemm_k(const float* __restrict__ X,
                                                   const float* __restrict__ W,
                                                   const float* __restrict__ v,
                                                   float* __restrict__ blog) {
    int bid  = blockIdx.x;
    int bc   = bid % NBCHUNKS;                 // batch chunk
    int mt   = bid / NBCHUNKS;                 // 0..575 (ik tile)
    int lane = threadIdx.x;
    int half = lane >> 4;
    int ln   = lane & 15;
    int ik0  = mt * 16;
    int kb0  = bc * BCHUNK;

    v8f acc[CAP_J] = {};
    const float* xcol = X + (size_t)kb0 * CAP_IK + ik0 + ln; // stride IK over batch
    const float* vrow = v + (size_t)kb0 * CAP_N;
    for (int ks = 0; ks < BCHUNK / 4; ++ks) {  // 16 wmma steps over this chunk
        int kb = 4 * ks + 2 * half;
        v2f a;
        a.x = xcol[(size_t)(kb + 0) * CAP_IK];
        a.y = xcol[(size_t)(kb + 1) * CAP_IK];
        #pragma unroll
        for (int j = 0; j < CAP_J; ++j) {      // reuse A across all output capsules
            v2f bb;
            bb.x = vrow[(size_t)(kb + 0) * CAP_N + j * CAP_D + ln];
            bb.y = vrow[(size_t)(kb + 1) * CAP_N + j * CAP_D + ln];
            acc[j] = __builtin_amdgcn_wmma_f32_16x16x4_f32(
                         false, a, false, bb, (short)0, acc[j], false, false);
        }
    }
    // D tile element: T[ik0 + r + 8*half][d=ln]; i = ik0/8 + half, k = r
    int i = (ik0 >> 3) + half;
    #pragma unroll
    for (int j = 0; j < CAP_J; ++j) {
        const float* wp = W + (size_t)(i * CAP_J + j) * (CAP_DIN * CAP_D) + ln;
        float partial = 0.f;
        #pragma unroll
        for (int r = 0; r < 8; ++r) partial += acc[j][r] * wp[r * CAP_D];
        // reduce over the 16 lanes of this half (sum over k via regs, d via lanes)
        partial += __shfl_xor(partial, 1);
        partial += __shfl_xor(partial, 2);
        partial += __shfl_xor(partial, 4);
        partial += __shfl_xor(partial, 8);
        if (ln == 0) atomicAdd(&blog[i * CAP_J + j], partial);
    }
}

// ---------------------------------------------------------------------------
extern "C" void kernel_launch(void* const* d_in, const int* in_sizes, int n_in,
                              void* d_out, int out_size, void* d_ws, size_t ws_size,
                              hipStream_t stream) {
    const float* X    = (const float*)d_in[0];   // inputs [256,1152,8] == X'[256,9216]
    const float* W    = (const float*)d_in[1];   // [1152*10, 8, 16]
    const float* bias = (const float*)d_in[2];   // [1152,10]
    float* out = (float*)d_out;                  // v [256,10,16]
    float* ws  = (float*)d_ws;

    float* blog = ws;                            // 11520 f
    float* c    = ws + 11520;                    // 11520 f
    float* s    = ws + 23040;                    // 40960 f
    float* vbuf = ws + 64000;                    // 40960 f

    hipMemcpyAsync(blog, bias, (size_t)CAP_I * CAP_J * sizeof(float),
                   hipMemcpyDeviceToDevice, stream);

    for (int r = 0; r < 3; ++r) {
        softmax_k<<<(CAP_I + 127) / 128, 128, 0, stream>>>(blog, c);
        hipMemsetAsync(s, 0, (size_t)CAP_B * CAP_N * sizeof(float), stream);
        s_gemm_k<<<16 * NCHUNKS, 32, 0, stream>>>(X, W, c, s);
        float* vdst = (r == 2) ? out : vbuf;
        squash_k<<<(CAP_B * CAP_J + 255) / 256, 256, 0, stream>>>(s, vdst);
        if (r != 2)
            agree_gemm_k<<<(CAP_IK / 16) * NBCHUNKS, 32, 0, stream>>>(X, W, vbuf, blog);
    }
}